// LSTM_13615046328915
// MI455X (gfx1250) — compile-verified
//
#include <hip/hip_runtime.h>
#include <hip/hip_bf16.h>
#include <math.h>
#include <stdint.h>

// LSTM LM: V=32000, E=H=1024, LAYERS=2, unbatched T=512 scan + FC head.
// gather -> [per layer: fp32-WMMA GEMM input projection, persistent-scan kernel]
// -> fp32-WMMA GEMM FC head -> state copy.  GEMMs use double-buffered LDS fed by
// global_load_async_to_lds_b128 (ASYNCcnt pipeline).

typedef __attribute__((ext_vector_type(2))) float v2f;
typedef __attribute__((ext_vector_type(8))) float v8f;

#define T_STEPS 512
#define EDIM 1024
#define HDIM 1024
#define G4 4096
#define VOCAB 32000
#define CK 32            // K-chunk staged per LDS buffer
#define LDSROW 36        // 32 + 4 pad floats (bank spread, keeps 16B align)

__device__ __forceinline__ float sigmoidf_(float x) { return 1.0f / (1.0f + __expf(-x)); }

// CDNA5 async copy: global -> LDS, tracked by ASYNCcnt (ISA 10.x / 15.18.3)
__device__ __forceinline__ void async_b128(uint32_t lds_off, const float* g) {
  asm volatile("global_load_async_to_lds_b128 %0, %1, off"
               :: "v"(lds_off), "v"((unsigned long long)(uintptr_t)g)
               : "memory");
}

// ---------------- embedding gather (time-major seq) -----------------
// seq[l*N + n] = emb[x[n][l]],  N=8, L=64
__global__ void gather_kernel(const int* __restrict__ x, const float* __restrict__ emb,
                              float* __restrict__ seq) {
  int s = blockIdx.x;
  int n = s & 7;
  int l = s >> 3;
  int idx = x[n * 64 + l];
  const float4* src = (const float4*)(emb + (size_t)idx * EDIM);
  float4* dst = (float4*)(seq + (size_t)s * EDIM);
  dst[threadIdx.x] = src[threadIdx.x];
}

__global__ void init_kernel(unsigned* __restrict__ bar) {
  if (threadIdx.x < 2) bar[threadIdx.x] = 0u;
}

// ---------------- fp32 WMMA GEMM: out = A(MxK) * W(NoutxK)^T + bias1 + bias2 ---
// 256 threads = 8 waves. Block tile 32(M) x 128(N); wave owns a 32x16 C slab
// (two v8f accumulators sharing each B fragment). K staged through ping-pong
// LDS buffers filled by async-to-LDS b128 loads; s_wait_asynccnt pipelines.
__global__ void wmma_gemm_bias(const float* __restrict__ A,
                               const float* __restrict__ W,
                               const float* __restrict__ bias1,
                               const float* __restrict__ bias2,
                               float* __restrict__ out,
                               int M, int K, int Nout) {
  __shared__ float As[2][32 * LDSROW];
  __shared__ float Ws[2][128 * LDSROW];
  const int tid = threadIdx.x;
  const int wave = tid >> 5;
  const int lane = tid & 31;
  const int colBase = blockIdx.x * 128;
  const int rowBase = blockIdx.y * 32;

  const int nGlob = colBase + wave * 16 + (lane & 15);
  float binit = 0.0f;
  if (bias1) binit += bias1[nGlob];
  if (bias2) binit += bias2[nGlob];
  v8f acc0, acc1;
#pragma unroll
  for (int r = 0; r < 8; ++r) { acc0[r] = binit; acc1[r] = binit; }

  const int mlocal = lane & 15;
  const int koff = (lane < 16) ? 0 : 2;      // A/B fragment layout, ISA 7.12.2
  const int nLocal = wave * 16 + (lane & 15);

  // per-thread staging coordinates: 32x32 A tile (256 xfer) + 128x32 W tile (1024)
  const int aRow = tid >> 3;                 // 0..31
  const int aC4 = tid & 7;                   // 0..7 (float4 column)
  uint32_t asBase[2], wsBase[2];
  asBase[0] = (uint32_t)(uintptr_t)&As[0][0];
  asBase[1] = (uint32_t)(uintptr_t)&As[1][0];
  wsBase[0] = (uint32_t)(uintptr_t)&Ws[0][0];
  wsBase[1] = (uint32_t)(uintptr_t)&Ws[1][0];
  const float* Ag = A + (size_t)(rowBase + aRow) * K + aC4 * 4;
  const float* Wg = W + (size_t)(colBase + aRow) * K + aC4 * 4;

  auto issue = [&](int s, int k0) {
    async_b128(asBase[s] + (uint32_t)(aRow * LDSROW + aC4 * 4) * 4, Ag + k0);
#pragma unroll
    for (int j = 0; j < 4; ++j)
      async_b128(wsBase[s] + (uint32_t)((j * 32 + aRow) * LDSROW + aC4 * 4) * 4,
                 Wg + (size_t)j * 32 * K + k0);
  };

  issue(0, 0);
  int cur = 0;
  for (int k0 = 0; k0 < K; k0 += CK, cur ^= 1) {
    if (k0 + 2 * CK < K)   // stream W toward L2 ahead of the async pipeline
      __builtin_prefetch(Wg + k0 + 2 * CK, 0, 1);
    if (k0 + CK < K) {
      issue(cur ^ 1, k0 + CK);                      // 5 more in flight (10 total)
      asm volatile("s_wait_asynccnt %0" ::"i"(5) : "memory");  // cur chunk landed
    } else {
      asm volatile("s_wait_asynccnt %0" ::"i"(0) : "memory");
    }
    __syncthreads();
    const float* as = &As[cur][0];
    const float* wsp = &Ws[cur][0];
#pragma unroll
    for (int kk = 0; kk < CK / 4; ++kk) {
      v2f a0 = *(const v2f*)(&as[mlocal * LDSROW + kk * 4 + koff]);
      v2f a1 = *(const v2f*)(&as[(16 + mlocal) * LDSROW + kk * 4 + koff]);
      v2f b = *(const v2f*)(&wsp[nLocal * LDSROW + kk * 4 + koff]);
      acc0 = __builtin_amdgcn_wmma_f32_16x16x4_f32(false, a0, false, b, (short)0,
                                                   acc0, false, false);
      acc1 = __builtin_amdgcn_wmma_f32_16x16x4_f32(false, a1, false, b, (short)0,
                                                   acc1, false, false);
    }
    __syncthreads();
  }

#pragma unroll
  for (int r = 0; r < 8; ++r) {
    int m = rowBase + r + ((lane >> 4) << 3);
    out[(size_t)m * Nout + nGlob] = acc0[r];
    out[(size_t)(m + 16) * Nout + nGlob] = acc1[r];
  }
}

// ---------------- persistent recurrent scan for one layer ---------------------
// 64 blocks x 256 threads stay resident for all 512 steps; agent-scope atomic
// grid barrier between steps. Block b owns hidden units [b*16, b*16+16) -> 64
// gate rows; each of 8 waves does 8 row-dots of length 1024. Cell state lives
// in registers of threads 0..15.
__device__ __forceinline__ void grid_sync(unsigned* bar, unsigned target) {
  __threadfence();          // make this block's h stores device-visible
  __syncthreads();
  if (threadIdx.x == 0) {
    __hip_atomic_fetch_add(bar, 1u, __ATOMIC_ACQ_REL, __HIP_MEMORY_SCOPE_AGENT);
    while (__hip_atomic_load(bar, __ATOMIC_ACQUIRE, __HIP_MEMORY_SCOPE_AGENT) < target)
      __builtin_amdgcn_s_sleep(1);
  }
  __syncthreads();
}

__global__ void lstm_scan_kernel(const float* __restrict__ Xg,   // (T,4H) preproj
                                 const float* __restrict__ Whh,  // (4H,H)
                                 float* __restrict__ outH,       // (T,H)
                                 float* __restrict__ cFinal,     // (H)
                                 unsigned* __restrict__ bar) {
  __shared__ float gs[64];
  const int b = blockIdx.x;
  const int tid = threadIdx.x;
  const int wave = tid >> 5;
  const int lane = tid & 31;
  float cR = 0.0f;

  for (int t = 0; t < T_STEPS; ++t) {
    float4 hreg[8];
#pragma unroll
    for (int i = 0; i < 8; ++i) hreg[i] = make_float4(0.f, 0.f, 0.f, 0.f);
    if (t > 0) {
      const float* hprev = outH + (size_t)(t - 1) * HDIM;
#pragma unroll
      for (int i = 0; i < 8; ++i)
        hreg[i] = *(const float4*)(hprev + (i * 32 + lane) * 4);
    }

#pragma unroll
    for (int rr = 0; rr < 8; ++rr) {
      int rowIdx = wave * 8 + rr;                // 0..63
      int q = rowIdx >> 4;                       // gate: i,f,g,o
      int u = rowIdx & 15;
      float red = 0.0f;
      if (t > 0) {
        const float* wrow = Whh + (size_t)(q * HDIM + b * 16 + u) * HDIM;
#pragma unroll
        for (int i = 0; i < 8; ++i) {
          float4 w4 = *(const float4*)(wrow + (i * 32 + lane) * 4);
          red = fmaf(w4.x, hreg[i].x, red);
          red = fmaf(w4.y, hreg[i].y, red);
          red = fmaf(w4.z, hreg[i].z, red);
          red = fmaf(w4.w, hreg[i].w, red);
        }
        for (int off = 16; off > 0; off >>= 1) red += __shfl_down(red, off, 32);
      }
      if (lane == 0) gs[rowIdx] = red;
    }
    __syncthreads();

    if (tid < 16) {
      int gu = b * 16 + tid;
      const float* xg = Xg + (size_t)t * G4;
      float vi = xg[0 * HDIM + gu] + gs[0 * 16 + tid];
      float vf = xg[1 * HDIM + gu] + gs[1 * 16 + tid];
      float vg = xg[2 * HDIM + gu] + gs[2 * 16 + tid];
      float vo = xg[3 * HDIM + gu] + gs[3 * 16 + tid];
      float i_ = sigmoidf_(vi);
      float f_ = sigmoidf_(vf);
      float g_ = tanhf(vg);
      float o_ = sigmoidf_(vo);
      cR = f_ * cR + i_ * g_;
      outH[(size_t)t * HDIM + gu] = o_ * tanhf(cR);
    }

    if (t + 1 < T_STEPS) grid_sync(bar, 64u * (unsigned)(t + 1));
  }

  if (tid < 16) cFinal[b * 16 + tid] = cR;
}

// ---------------- final (h, c) states into output tail ------------------------
__global__ void finalize_kernel(const float* __restrict__ o0, const float* __restrict__ o1,
                                const float* __restrict__ c0, const float* __restrict__ c1,
                                float* __restrict__ tail) {
  int l = blockIdx.x;
  int j = threadIdx.x;
  const float* oh = l ? o1 : o0;
  const float* cc = l ? c1 : c0;
  tail[l * HDIM + j] = oh[(size_t)(T_STEPS - 1) * HDIM + j];   // h (2,1024)
  tail[2 * HDIM + l * HDIM + j] = cc[j];                       // c (2,1024)
}

extern "C" void kernel_launch(void* const* d_in, const int* in_sizes, int n_in,
                              void* d_out, int out_size, void* d_ws, size_t ws_size,
                              hipStream_t stream) {
  const int*   x    = (const int*)d_in[0];
  const float* emb  = (const float*)d_in[1];
  const float* W_ih = (const float*)d_in[2];   // (2, 4096, 1024)
  const float* W_hh = (const float*)d_in[3];   // (2, 4096, 1024)
  const float* b_ih = (const float*)d_in[4];   // (2, 4096)
  const float* b_hh = (const float*)d_in[5];   // (2, 4096)
  const float* W_fc = (const float*)d_in[6];   // (32000, 1024)
  const float* b_fc = (const float*)d_in[7];   // (32000)
  float* out = (float*)d_out;

  float* ws   = (float*)d_ws;
  float* seq  = ws;                              // 512*1024
  float* Xg   = seq  + (size_t)T_STEPS * EDIM;   // 512*4096
  float* out0 = Xg   + (size_t)T_STEPS * G4;     // 512*1024
  float* out1 = out0 + (size_t)T_STEPS * HDIM;   // 512*1024
  float* c0   = out1 + (size_t)T_STEPS * HDIM;   // 1024
  float* c1   = c0 + HDIM;                       // 1024
  unsigned* bar = (unsigned*)(c1 + HDIM);        // 2 grid-barrier counters

  init_kernel<<<1, 32, 0, stream>>>(bar);
  gather_kernel<<<T_STEPS, 256, 0, stream>>>(x, emb, seq);

  // ----- layer 0 -----
  wmma_gemm_bias<<<dim3(G4 / 128, T_STEPS / 32), 256, 0, stream>>>(
      seq, W_ih, b_ih, b_hh, Xg, T_STEPS, EDIM, G4);
  lstm_scan_kernel<<<64, 256, 0, stream>>>(Xg, W_hh, out0, c0, bar + 0);

  // ----- layer 1 -----
  wmma_gemm_bias<<<dim3(G4 / 128, T_STEPS / 32), 256, 0, stream>>>(
      out0, W_ih + (size_t)G4 * EDIM, b_ih + G4, b_hh + G4, Xg, T_STEPS, HDIM, G4);
  lstm_scan_kernel<<<64, 256, 0, stream>>>(Xg, W_hh + (size_t)G4 * HDIM, out1, c1, bar + 1);

  // ----- FC head: scores into d_out[0 : 512*32000) -----
  wmma_gemm_bias<<<dim3(VOCAB / 128, T_STEPS / 32), 256, 0, stream>>>(
      out1, W_fc, b_fc, nullptr, out, T_STEPS, HDIM, VOCAB);

  finalize_kernel<<<2, 1024, 0, stream>>>(out0, out1, c0, c1,
                                          out + (size_t)T_STEPS * VOCAB);
}